// MLPPredictor_7318624272588
// MI455X (gfx1250) — compile-verified
//
#include <hip/hip_runtime.h>

typedef float v2f __attribute__((ext_vector_type(2)));
typedef float v8f __attribute__((ext_vector_type(8)));

#define N_EDGES 500000
#define D_FEAT 128
#define OUT_C 64
#define EPC 32                       /* edges per chunk */
#define NCHUNKS (N_EDGES / EPC)      /* 15625, exact */
#define PAD 268                      /* floats per staged edge row (256+12): conflict-free, 16B aligned */
#define BUF_FLOATS (EPC * PAD)
#define NUM_WG 2048

// Per-lane async copy of 16B global -> LDS (ASYNCcnt, no data VGPRs).
// GVS addressing: mem = SGPR64(hbase) + VGPR32(byte offset).
__device__ __forceinline__ void async_gather_b128(unsigned lds_off, unsigned gofs,
                                                  const float* hbase) {
  asm volatile("global_load_async_to_lds_b128 %0, %1, %2"
               :
               : "v"(lds_off), "v"(gofs), "s"(hbase)
               : "memory");
}
__device__ __forceinline__ void wait_async0() {
  asm volatile("s_wait_asynccnt 0x0" ::: "memory");
}

__global__ __launch_bounds__(256) void gnn_edge_score_kernel(
    const float* __restrict__ h, const int* __restrict__ src,
    const int* __restrict__ dst, const float* __restrict__ W,
    const float* __restrict__ b, float* __restrict__ out) {
  extern __shared__ float feat[];  // 2 * BUF_FLOATS (double buffer)

  const int tid  = threadIdx.x;
  const int lane = tid & 31;
  const int wave = tid >> 5;   // 0..7
  const int ct   = wave & 3;   // class tile
  const int et   = wave >> 2;  // edge tile
  const int col  = lane & 15;  // A: M row, B/C: N column
  const int half = lane >> 4;  // K half within 4-wide step

  // ---- B fragments (W^T tile) in VGPRs, loaded once ----
  v2f bfrag[64];
  {
    const float* wrow = W + (size_t)(ct * 16 + col) * (2 * D_FEAT) + 2 * half;
#pragma unroll
    for (int kk = 0; kk < 64; ++kk) bfrag[kk] = *(const v2f*)(wrow + 4 * kk);
  }
  const float bias = b[ct * 16 + col];

  // ---- staging geometry: slot v = i*256 + tid ----
  // edge = 4*i + (tid>>6), part = (tid>>5)&1, off = tid&31 -> per-thread constants.
  // Each wave's 32 lanes cover one contiguous 512B h-row per i.
  const int s_c    = tid >> 6;
  const int s_part = (tid >> 5) & 1;
  const unsigned s_ofsB = (unsigned)(tid & 31) * 16u;  // byte offset within h row
  const int* __restrict__ pidx = s_part ? dst : src;
  const int lds_sbase = s_c * PAD + s_part * 128 + (tid & 31) * 4;  // floats; + i*(4*PAD)

  int chunk = blockIdx.x;
  if (chunk >= NCHUNKS) return;
  const int gstride = (int)gridDim.x;

  // ---- prologue: async-stage first chunk into buffer 0 ----
  {
    const int e0 = chunk * EPC;
#pragma unroll
    for (int i = 0; i < 8; ++i) {
      unsigned idx = (unsigned)pidx[e0 + s_c + 4 * i];
      unsigned ldso = (unsigned)(size_t)(feat + lds_sbase + i * (4 * PAD));
      async_gather_b128(ldso, (idx << 9) + s_ofsB, h);
    }
  }
  // prefetch indices for the next chunk while the asyncs fly
  int nchunk = chunk + gstride;
  int nidx[8];
  if (nchunk < NCHUNKS) {
    const int e0n = nchunk * EPC;
#pragma unroll
    for (int i = 0; i < 8; ++i) nidx[i] = pidx[e0n + s_c + 4 * i];
  }
  wait_async0();
  __syncthreads();

  int buf = 0;
  while (true) {
    const bool have_next = (nchunk < NCHUNKS);

    // ---- issue next chunk's async gathers into the other buffer ----
    if (have_next) {
      float* base = feat + (buf ^ 1) * BUF_FLOATS + lds_sbase;
#pragma unroll
      for (int i = 0; i < 8; ++i) {
        unsigned gofs = ((unsigned)nidx[i] << 9) + s_ofsB;
        async_gather_b128((unsigned)(size_t)(base + i * (4 * PAD)), gofs, h);
      }
    }
    // ---- prefetch indices for the chunk after next (hidden behind WMMAs) ----
    const int n2 = nchunk + gstride;
    if (n2 < NCHUNKS) {
      const int e0n2 = n2 * EPC;
#pragma unroll
      for (int i = 0; i < 8; ++i) nidx[i] = pidx[e0n2 + s_c + 4 * i];
    }

    // ---- 16x16 tile: 64 x v_wmma_f32_16x16x4_f32 over K=256 ----
    v8f acc = {};
    const float* fb = feat + buf * BUF_FLOATS + (et * 16 + col) * PAD + 2 * half;
#pragma unroll
    for (int kk = 0; kk < 64; ++kk) {
      v2f a = *(const v2f*)(fb + 4 * kk);
      acc = __builtin_amdgcn_wmma_f32_16x16x4_f32(
          /*neg_a=*/false, a, /*neg_b=*/false, bfrag[kk],
          /*c_mod=*/(short)0, acc, /*reuse_a=*/false, /*reuse_b=*/false);
    }

    // ---- non-temporal output stores: keep h/W resident in L2 ----
    const int e0 = chunk * EPC;
#pragma unroll
    for (int j = 0; j < 8; ++j) {
      int edge = e0 + et * 16 + (half ? (j + 8) : j);
      __builtin_nontemporal_store(acc[j] + bias,
                                  &out[(size_t)edge * OUT_C + ct * 16 + col]);
    }

    if (!have_next) break;

    wait_async0();      // our async writes into buf^1 are complete
    __syncthreads();    // everyone's are; safe to read next iteration
    chunk = nchunk;
    nchunk = n2;
    buf ^= 1;
  }
}

extern "C" void kernel_launch(void* const* d_in, const int* in_sizes, int n_in,
                              void* d_out, int out_size, void* d_ws, size_t ws_size,
                              hipStream_t stream) {
  const float* h   = (const float*)d_in[0];
  const int*   src = (const int*)d_in[1];
  const int*   dst = (const int*)d_in[2];
  const float* W   = (const float*)d_in[3];
  const float* b   = (const float*)d_in[4];
  float* out = (float*)d_out;
  gnn_edge_score_kernel<<<NUM_WG, 256, 2 * BUF_FLOATS * (int)sizeof(float), stream>>>(
      h, src, dst, W, b, out);
}